// PhysicsEngine_42202348650870
// MI455X (gfx1250) — compile-verified
//
#include <hip/hip_runtime.h>
#include <math.h>

typedef __attribute__((ext_vector_type(2))) float v2f;
typedef __attribute__((ext_vector_type(8))) float v8f;

#define BB    8      // batch
#define NL    128    // ligand atoms
#define NP    8192   // protein atoms
#define NPT   (NP / 16)   // 512 P-tiles per batch
#define NWAVES 8

__device__ __forceinline__ float fast_rcp(float x)  { return __builtin_amdgcn_rcpf(x); }
__device__ __forceinline__ float fast_sqrt(float x) { return __builtin_amdgcn_sqrtf(x); }

__device__ __forceinline__ float softplus_f(float x) {
    // stable softplus with hardware transcendentals:
    // softplus(x) = max(x,0) + log(1 + exp(-|x|)); exp(-|x|) in (0,1] so the
    // log(1+u) form loses at most ~6e-8 absolute vs log1p -- negligible here.
    float u = __expf(-fabsf(x));
    return fmaxf(x, 0.0f) + __logf(1.0f + u);
}

// ---------------------------------------------------------------------------
// Kernel 0: ligand-side per-atom precompute into workspace
//   rl   = x_L @ vdw_radii
//   epsT = sqrt(relu(x_L @ epsilon) * 0.15 + 1e-8)
//   cL   = x_L[..., 0]
//   nL   = |pos_L|^2
// ---------------------------------------------------------------------------
__global__ void precomp_ligand(const float* __restrict__ posL,
                               const float* __restrict__ xL,
                               const float* __restrict__ vdw,
                               const float* __restrict__ eps,
                               float* __restrict__ rl,
                               float* __restrict__ epsT,
                               float* __restrict__ cL,
                               float* __restrict__ nL) {
    int a = blockIdx.x * blockDim.x + threadIdx.x;
    if (a >= BB * NL) return;
    float r = 0.0f, e = 0.0f;
#pragma unroll
    for (int j = 0; j < 9; ++j) {
        float x = xL[a * 9 + j];
        r += x * vdw[j];
        e += x * eps[j];
    }
    rl[a]   = r;
    epsT[a] = sqrtf(fmaxf(e, 0.0f) * 0.15f + 1e-8f);
    cL[a]   = xL[a * 9];
    float px = posL[a * 3], py = posL[a * 3 + 1], pz = posL[a * 3 + 2];
    nL[a] = px * px + py * py + pz * pz;
}

// ---------------------------------------------------------------------------
// Kernel 1: tiled all-pairs energy. One block = one batch b x one P-tile (16
// cols); wave w (of 8) = L-tile rows [16w, 16w+16). Each wave computes the
// 16x16 tile of L.P^T with V_WMMA_F32_16X16X4_F32 (K=3 padded), then per-lane
// scalar energy math over its 8 C-matrix elements.
// ---------------------------------------------------------------------------
__global__ void __launch_bounds__(256)
pair_energy(const float* __restrict__ posL, const float* __restrict__ posP,
            const float* __restrict__ qL,   const float* __restrict__ qP,
            const float* __restrict__ xP,
            const float* __restrict__ rl,   const float* __restrict__ epsT,
            const float* __restrict__ cL,   const float* __restrict__ nL,
            float* __restrict__ acc) {
    __shared__ float sPx[16], sPy[16], sPz[16];
    __shared__ float sRp[16], sNp[16], sQp[16], sCp[16];
    __shared__ float red[NWAVES][4];

    const int b   = blockIdx.x / NPT;
    const int tn  = blockIdx.x % NPT;
    const int tid = threadIdx.x;

    // stage the 16 protein atoms of this tile into LDS
    if (tid < 16) {
        int n = b * NP + tn * 16 + tid;
        float px = posP[n * 3], py = posP[n * 3 + 1], pz = posP[n * 3 + 2];
        sPx[tid] = px; sPy[tid] = py; sPz[tid] = pz;
        sNp[tid] = px * px + py * py + pz * pz;
        sQp[tid] = qP[n];
        float x0 = xP[n * 4], x1 = xP[n * 4 + 1];
        float x2 = xP[n * 4 + 2], x3 = xP[n * 4 + 3];
        sRp[tid] = 1.70f * x0 + 1.55f * x1 + 1.52f * x2 + 1.80f * x3;
        sCp[tid] = x0;
    }
    __syncthreads();

    const int  w    = tid >> 5;      // wave id == L tile index
    const int  lane = tid & 31;
    const int  nloc = lane & 15;
    const bool hi   = lane >= 16;

    // A fragment (16x4 f32): lanes 0-15 hold {K0,K1}={x,y} for M=lane,
    //                        lanes 16-31 hold {K2,K3}={z,0} for M=lane-16
    int lrow = b * NL + w * 16 + nloc;
    float lx = posL[lrow * 3], ly = posL[lrow * 3 + 1], lz = posL[lrow * 3 + 2];
    v2f afrag; afrag.x = hi ? lz : lx;        afrag.y = hi ? 0.0f : ly;
    // B fragment (4x16 f32): same K striping, N = lane&15
    v2f bfrag; bfrag.x = hi ? sPz[nloc] : sPx[nloc];
               bfrag.y = hi ? 0.0f      : sPy[nloc];

    v8f cacc = {};
    cacc = __builtin_amdgcn_wmma_f32_16x16x4_f32(
        /*neg_a=*/false, afrag, /*neg_b=*/false, bfrag,
        /*c_mod=*/(short)0, cacc, /*reuse_a=*/false, /*reuse_b=*/false);
    // cacc[i] = dot(L_{m}, P_{n}) with m = i + 8*hi, n = nloc

    const float qp_n = sQp[nloc], rp_n = sRp[nloc];
    const float np_n = sNp[nloc], cp_n = sCp[nloc];
    const int mbase = b * NL + w * 16 + (hi ? 8 : 0);

    float s1 = 0.0f, s2 = 0.0f, s3 = 0.0f, s4 = 0.0f;
#pragma unroll
    for (int i = 0; i < 8; ++i) {
        int   mg   = mbase + i;
        float dot  = cacc[i];
        float d2   = fmaxf(nL[mg] + np_n - 2.0f * dot, 0.0f);
        float dist = fast_sqrt(d2 + 1e-8f);
        float sig  = rl[mg] + rp_n;
        float soft = fast_sqrt(d2 + sig * sig + 1e-8f);   // alpha = 1
        float e_elec = 83.015f * qL[mg] * qp_n * fast_rcp(soft);  // 332.06 / 4
        float ratio  = fminf(sig * fast_rcp(soft + 1e-8f), 5.0f);
        float r2 = ratio * ratio, r6 = r2 * r2 * r2, r12 = r6 * r6;
        float evraw = 4.0f * epsT[mg] * (r12 - r6);
        float v = softplus_f(evraw + 10.0f) - 10.0f;      // soft clamp low
        v = 500.0f - softplus_f(500.0f - v);              // soft clamp high
        float mask = fast_rcp(1.0f + __expf(2.0f * dist - 24.0f)); // sigmoid((12-d)*2)
        s1 += (e_elec + v) * mask;
        float t = dist * 0.25f; float t2 = t * t;
        s2 -= 0.5f * cL[mg] * cp_n * mask * fast_rcp(1.0f + t2 * t2);
        float ov = fmaxf(sig * 0.6f - dist, 0.0f);
        float gh = fmaxf(0.5f - dist, 0.0f);
        s3 += 11.920292f * ov * ov + 500.0f * gh * gh;    // 100*sigmoid(-2), ghost
        float lv = fminf(fmaxf(evraw, -10.0f), 500.0f);
        s4 += (e_elec + lv) * mask;
    }

    // wave32 reduction
#pragma unroll
    for (int off = 16; off >= 1; off >>= 1) {
        s1 += __shfl_down(s1, off);
        s2 += __shfl_down(s2, off);
        s3 += __shfl_down(s3, off);
        s4 += __shfl_down(s4, off);
    }
    if (lane == 0) { red[w][0] = s1; red[w][1] = s2; red[w][2] = s3; red[w][3] = s4; }
    __syncthreads();
    if (tid < 4) {
        float v = 0.0f;
#pragma unroll
        for (int ww = 0; ww < NWAVES; ++ww) v += red[ww][tid];
        atomicAdd(&acc[tid * BB + b], v);
    }
}

// ---------------------------------------------------------------------------
// Kernel 2: finalize the three outputs per batch
// ---------------------------------------------------------------------------
__global__ void finalize(const float* __restrict__ acc, float* __restrict__ out) {
    int b = threadIdx.x;
    if (b >= BB) return;
    float S1 = acc[0 * BB + b];   // sum (e_elec + e_vdw_grad)*mask
    float S2 = acc[1 * BB + b];   // e_hsa
    float S3 = acc[2 * BB + b];   // e_pauli + e_ghost
    float S4 = acc[3 * BB + b];   // sum (e_elec + clipped vdw)*mask
    float e_raw  = S1 + 5.0f * S2 + S3;
    float e_hard = fminf(S3, 10000.0f);
    float lsoft  = S4 + 5.0f * S2;
    float esf    = fminf(fmaxf(lsoft, -500.0f), 5000.0f);
    float le     = fminf(esf + e_hard, 1000000.0f);
    out[b]          = e_raw;
    out[BB + b]     = e_hard;
    out[2 * BB + b] = le;
}

extern "C" void kernel_launch(void* const* d_in, const int* in_sizes, int n_in,
                              void* d_out, int out_size, void* d_ws, size_t ws_size,
                              hipStream_t stream) {
    const float* posL = (const float*)d_in[0];
    const float* posP = (const float*)d_in[1];
    const float* qL   = (const float*)d_in[2];
    const float* qP   = (const float*)d_in[3];
    const float* xL   = (const float*)d_in[4];
    const float* xP   = (const float*)d_in[5];
    const float* vdw  = (const float*)d_in[6];
    const float* eps  = (const float*)d_in[7];

    float* ws   = (float*)d_ws;
    float* acc  = ws;                 // 4*BB floats
    float* rl   = ws + 4 * BB;        // BB*NL
    float* epsT = rl + BB * NL;       // BB*NL
    float* cL   = epsT + BB * NL;     // BB*NL
    float* nL   = cL + BB * NL;       // BB*NL

    hipMemsetAsync(acc, 0, 4 * BB * sizeof(float), stream);
    precomp_ligand<<<(BB * NL + 255) / 256, 256, 0, stream>>>(
        posL, xL, vdw, eps, rl, epsT, cL, nL);
    pair_energy<<<BB * NPT, 256, 0, stream>>>(
        posL, posP, qL, qP, xP, rl, epsT, cL, nL, acc);
    finalize<<<1, 32, 0, stream>>>(acc, (float*)d_out);
}